// SphericalHarmonics_6399501271671
// MI455X (gfx1250) — compile-verified
//
#include <hip/hip_runtime.h>
#include <math.h>

// ---------------------------------------------------------------------------
// Spherical harmonics Y_lm, L=10 (100 cols), N=1e6 points, fp32.
// Store-bandwidth-bound: 400MB out @ 23.3TB/s ~= 17us floor; VALU math ~3us.
// Strategy: direct unrolled Legendre recurrence -> LDS staging -> async
// LDS->global b128 stores (CDNA5 async data mover, ASYNCcnt).
// ---------------------------------------------------------------------------

#define LP  10
#define TPB 128   // 128 pts/block -> 51,200B LDS staging = block's contiguous output

// ---- compile-time K_lm table (constexpr sqrt: Newton, safe for tiny args) --
constexpr double csqrt(double x) {
  if (x <= 0.0) return 0.0;
  double g = (x > 1.0) ? x : 1.0;
  for (int i = 0; i < 80; ++i) g = 0.5 * (g + x / g);
  return g;
}
constexpr double dfact(int n) {
  double r = 1.0;
  for (int i = 2; i <= n; ++i) r *= (double)i;
  return r;
}
struct KTab { float v[LP][LP]; };
constexpr KTab mkK() {
  KTab t{};
  for (int l = 0; l < LP; ++l)
    for (int m = 0; m <= l; ++m) {
      double K = csqrt((2.0 * l + 1.0) / (4.0 * 3.14159265358979323846)
                       * dfact(l - m) / dfact(l + m));
      // bake sqrt(2) into m>0 entries (used by both +m and -m columns)
      t.v[l][m] = (float)((m > 0) ? K * 1.4142135623730950488 : K);
    }
  return t;
}
constexpr KTab KT = mkK();

__global__ __launch_bounds__(TPB)
void sh_l10_kernel(const float2* __restrict__ lonlat, float* __restrict__ out, int N) {
  __shared__ __align__(16) float sOut[TPB * 100];

  const int tid        = threadIdx.x;
  const int blockStart = blockIdx.x * TPB;
  const int idx        = blockStart + tid;

  if (idx < N) {
    float2 p = lonlat[idx];                       // coalesced b64 load
    const float D2R = 0.017453292519943295f;
    float phi   = (p.x + 180.0f) * D2R;           // azimuth
    float theta = (p.y +  90.0f) * D2R;           // polar
    float s, x, sp, cp;
    sincosf(theta, &s,  &x);                      // s = sin(theta), x = cos(theta)
    sincosf(phi,   &sp, &cp);

    float* row = &sOut[tid * 100];

    float cm = 1.0f, sm = 0.0f;                   // cos(m*phi), sin(m*phi)
    float pmm = 1.0f;
#pragma unroll
    for (int m = 0; m < LP; ++m) {
      if (m > 0) {
        pmm *= -(2.0f * m - 1.0f) * s;            // Condon-Shortley phase
        float c2 = cm * cp - sm * sp;             // rotate by phi
        sm = sm * cp + cm * sp;
        cm = c2;
      }
      // P[m][m]
      float plm2 = pmm;
      {
        const int l = m; const float k = KT.v[l][m];
        if (m == 0) row[l * l + l] = k * plm2;
        else { row[l * l + l + m] = k * cm * plm2;
               row[l * l + l - m] = k * sm * plm2; }
      }
      if (m + 1 < LP) {
        // P[m+1][m]
        float plm1 = x * (2.0f * m + 1.0f) * pmm;
        {
          const int l = m + 1; const float k = KT.v[l][m];
          if (m == 0) row[l * l + l] = k * plm1;
          else { row[l * l + l + m] = k * cm * plm1;
                 row[l * l + l - m] = k * sm * plm1; }
        }
#pragma unroll
        for (int l = m + 2; l < LP; ++l) {
          float pl = ((2.0f * l - 1.0f) * x * plm1 - (float)(l + m - 1) * plm2)
                     * (1.0f / (float)(l - m));
          const float k = KT.v[l][m];
          if (m == 0) row[l * l + l] = k * pl;
          else { row[l * l + l + m] = k * cm * pl;
                 row[l * l + l - m] = k * sm * pl; }
          plm2 = plm1; plm1 = pl;
        }
      }
    }
  }
  __syncthreads();  // DS writes visible before async engine reads LDS

  // ---- drain LDS -> global via CDNA5 async store-from-LDS (b128/lane) ----
  // Block region is exactly contiguous in out[]: (blockStart..+TPB) x 100 f32.
  const int cnt         = (N - blockStart < TPB) ? (N - blockStart) : TPB;
  const int validFloats = cnt * 100;              // multiple of 4 -> clean 16B chunks
  float*   outBase = out + (size_t)blockStart * 100;
  unsigned ldsBase = (unsigned)(size_t)(void*)&sOut[0]; // low 32 bits = LDS offset

#pragma unroll 1
  for (int i = 0; i < 25; ++i) {                  // 25 * TPB * 16B = 51,200B
    int e = i * (TPB * 4) + tid * 4;              // float index in block region
    if (e < validFloats) {
      unsigned long long ga = (unsigned long long)(outBase + e);
      unsigned           la = ldsBase + (unsigned)e * 4u;
      asm volatile("global_store_async_from_lds_b128 %0, %1, off"
                   :: "v"(ga), "v"(la) : "memory");
    }
  }
  asm volatile("s_wait_asynccnt 0" ::: "memory"); // (S_ENDPGM also waits idle)
}

extern "C" void kernel_launch(void* const* d_in, const int* in_sizes, int n_in,
                              void* d_out, int out_size, void* d_ws, size_t ws_size,
                              hipStream_t stream) {
  (void)n_in; (void)out_size; (void)d_ws; (void)ws_size;
  const float2* lonlat = (const float2*)d_in[0];
  float*        out    = (float*)d_out;
  const int N      = in_sizes[0] / 2;             // in_sizes[0] = 2*N flat floats
  const int blocks = (N + TPB - 1) / TPB;
  sh_l10_kernel<<<blocks, TPB, 0, stream>>>(lonlat, out, N);
}